// STFTFCOSPostProcessor_21234318311580
// MI455X (gfx1250) — compile-verified
//
#include <hip/hip_runtime.h>
#include <math.h>

#define K 80
#define NLEV 3
#define TOPN 1000
#define CAP 4096           // per-level candidate capacity (power of 2 for bitonic)
#define NBINS 65536
#define NB 3072            // padded box count (3000 real)
#define MW 96              // NB/32 suppression mask words per row
#define SORT_N 4096

typedef __attribute__((ext_vector_type(2))) float v2f;
typedef __attribute__((ext_vector_type(8))) float v8f;

__device__ __forceinline__ float sigmoidf_(float x) { return 1.0f / (1.0f + __expf(-x)); }

// ---------------------------------------------------------------- init scratch
__global__ void init_ws_k(unsigned* hist, unsigned* candCount, float* candScore,
                          int* candIdx, unsigned* supMask) {
    int t = blockIdx.x * blockDim.x + threadIdx.x;
    int stride = gridDim.x * blockDim.x;
    for (int i = t; i < NLEV * NBINS; i += stride) hist[i] = 0u;
    for (int i = t; i < NLEV; i += stride) candCount[i] = 0u;
    for (int i = t; i < NLEV * CAP; i += stride) { candScore[i] = -1.0f; candIdx[i] = 0; }
    for (int i = t; i < NB * MW; i += stride) supMask[i] = 0u;
}

// -------------------------------------------------- fused scoring + histogram
// scoreBuf layout: [k*HW + p] (coalesced); ref flat index = p*K + k
__global__ void score_hist_k(const float* __restrict__ cls, const float* __restrict__ ctr,
                             const float* __restrict__ scls, float* __restrict__ scoreBuf,
                             unsigned* __restrict__ hist, int HW) {
    int t = blockIdx.x * blockDim.x + threadIdx.x;
    int total = HW * K;
    if (t >= total) return;
    int p = t % HW;
    float pc = sigmoidf_(cls[t]);
    float pt = sigmoidf_(ctr[p]);
    float ps = sigmoidf_(scls[t]);
    float prob = sqrtf(pc * pt);
    float m = (prob > 0.05f) ? prob * ps : -1.0f;
    scoreBuf[t] = m;
    if (m >= 0.0f) {
        int bin = (int)(m * (float)NBINS);
        if (bin > NBINS - 1) bin = NBINS - 1;
        atomicAdd(&hist[bin], 1u);
    }
}

// ------------------------------------------------- radix-select threshold bin
__global__ void find_thresh_k(const unsigned* __restrict__ hist, int* __restrict__ threshBin) {
    int lev = blockIdx.x;
    const unsigned* h = hist + lev * NBINS;
    __shared__ unsigned segSum[256];
    int tid = threadIdx.x;
    int base = NBINS - (tid + 1) * 256;      // segment tid counted from the top
    unsigned s = 0;
    for (int b = 0; b < 256; ++b) s += h[base + b];
    segSum[tid] = s;
    __syncthreads();
    if (tid == 0) {
        unsigned cum = 0; int seg = -1;
        for (int g = 0; g < 256; ++g) {
            if (cum + segSum[g] >= (unsigned)TOPN) { seg = g; break; }
            cum += segSum[g];
        }
        int bin = 0;
        if (seg >= 0) {
            int lo = NBINS - (seg + 1) * 256;
            bin = lo;
            for (int bb = NBINS - seg * 256 - 1; bb >= lo; --bb) {
                cum += h[bb];
                if (cum >= (unsigned)TOPN) { bin = bb; break; }
            }
        }
        threshBin[lev] = bin;
    }
}

// ----------------------------------------------------------- collect candidates
__global__ void collect_k(const float* __restrict__ scoreBuf, const int* __restrict__ threshBin,
                          unsigned* __restrict__ candCount, float* __restrict__ candScore,
                          int* __restrict__ candIdx, int HW, int lev) {
    int t = blockIdx.x * blockDim.x + threadIdx.x;
    int total = HW * K;
    if (t >= total) return;
    float s = scoreBuf[t];
    if (s < 0.0f) return;
    int bin = (int)(s * (float)NBINS);
    if (bin > NBINS - 1) bin = NBINS - 1;
    if (bin < threshBin[lev]) return;
    unsigned pos = atomicAdd(&candCount[lev], 1u);
    if (pos < CAP) {
        int p = t % HW, k = t / HW;
        candScore[lev * CAP + pos] = s;
        candIdx[lev * CAP + pos] = p * K + k;   // reference flat index
    }
}

// ---------------------------------------- bitonic sort 4096 (score desc, idx asc)
__global__ void bitonic4096_k(const float* __restrict__ key, const int* __restrict__ idx,
                              float* __restrict__ okey, int* __restrict__ oidx) {
    __shared__ float sk[SORT_N];
    __shared__ int   si[SORT_N];
    int tid = threadIdx.x;
    for (int i = tid; i < SORT_N; i += blockDim.x) { sk[i] = key[i]; si[i] = idx[i]; }
    __syncthreads();
    for (int k = 2; k <= SORT_N; k <<= 1) {
        for (int j = k >> 1; j > 0; j >>= 1) {
            for (int i = tid; i < SORT_N; i += blockDim.x) {
                int l = i ^ j;
                if (l > i) {
                    float ka = sk[i], kb = sk[l];
                    int ia = si[i], ib = si[l];
                    bool dir = ((i & k) == 0);                      // rank-ascending half
                    bool lBefore = (kb > ka) || (kb == ka && ib < ia);
                    if (lBefore == dir) { sk[i] = kb; sk[l] = ka; si[i] = ib; si[l] = ia; }
                }
            }
            __syncthreads();
        }
    }
    for (int i = tid; i < SORT_N; i += blockDim.x) { okey[i] = sk[i]; oidx[i] = si[i]; }
}

// --------------------------------------------------------------- box decoding
__global__ void decode_k(const float* __restrict__ selScore, const int* __restrict__ selIdx,
                         const float* sd0, const float* bx0, const float* sd1, const float* bx1,
                         const float* sd2, const float* bx2,
                         float* __restrict__ allBoxes, float* __restrict__ allScore,
                         int* __restrict__ allCls, int* __restrict__ allValid) {
    int r = blockIdx.x * blockDim.x + threadIdx.x;
    if (r >= NLEV * TOPN) return;
    int lev = r / TOPN, rank = r % TOPN;
    float sc = selScore[lev * CAP + rank];
    int idx = selIdx[lev * CAP + rank];
    const float* sd = (lev == 0) ? sd0 : (lev == 1) ? sd1 : sd2;
    const float* bx = (lev == 0) ? bx0 : (lev == 1) ? bx1 : bx2;
    int HW = (lev == 0) ? 67200 : (lev == 1) ? 16800 : 4200;
    int bi = idx / K;
    int ci = idx % K + 1;
    float x1 = bx[bi * 4 + 0], y1 = bx[bi * 4 + 1], x2 = bx[bi * 4 + 2], y2 = bx[bi * 4 + 3];
    float w = x2 - x1, h = y2 - y1;
    float d0 = sd[0 * HW + bi], d1 = sd[1 * HW + bi], d2 = sd[2 * HW + bi], d3 = sd[3 * HW + bi];
    allBoxes[r * 4 + 0] = x1 + d0 * 0.1f * w;
    allBoxes[r * 4 + 1] = y1 + d1 * 0.1f * h;
    allBoxes[r * 4 + 2] = x2 + d2 * 0.2f * w;
    allBoxes[r * 4 + 3] = y2 + d3 * 0.2f * h;
    allScore[r] = sqrtf(fmaxf(sc, 0.0f));
    allCls[r]   = ci;
    allValid[r] = (sc > 0.0f) ? 1 : 0;
}

// ------------------------------------------------ key build for the global sort
__global__ void build_gkey_k(const float* __restrict__ allScore, const int* __restrict__ allValid,
                             float* __restrict__ gkey, int* __restrict__ gidx) {
    int r = blockIdx.x * blockDim.x + threadIdx.x;
    if (r >= SORT_N) return;
    if (r < NLEV * TOPN) { gkey[r] = allValid[r] ? allScore[r] : -1.0f; gidx[r] = r; }
    else                 { gkey[r] = -2.0f;                              gidx[r] = r; }
}

// ------------------------------- gather in sorted order + max_coord + offsets
__global__ void gather_offset_k(const int* __restrict__ order, const float* __restrict__ allBoxes,
                                const float* __restrict__ allScore, const int* __restrict__ allCls,
                                const int* __restrict__ allValid,
                                float4* __restrict__ sBoxes, float* __restrict__ sScore,
                                int* __restrict__ sCls, int* __restrict__ sValid,
                                float4* __restrict__ offBoxes) {
    __shared__ float red[1024];
    __shared__ float sMax;
    int tid = threadIdx.x;
    float lm = 0.0f;
    for (int r = tid; r < NB; r += 1024) {
        float4 b = make_float4(0.f, 0.f, 0.f, 0.f);
        float sc = -1.0f; int c = 0, v = 0;
        if (r < NLEV * TOPN) {
            int o = order[r];
            b = make_float4(allBoxes[o * 4 + 0], allBoxes[o * 4 + 1],
                            allBoxes[o * 4 + 2], allBoxes[o * 4 + 3]);
            sc = allScore[o]; c = allCls[o]; v = allValid[o];
        }
        sBoxes[r] = b; sScore[r] = sc; sCls[r] = c; sValid[r] = v;
        if (v) lm = fmaxf(lm, fmaxf(fmaxf(b.x, b.y), fmaxf(b.z, b.w)));
    }
    red[tid] = lm;
    __syncthreads();
    for (int s = 512; s > 0; s >>= 1) {
        if (tid < s) red[tid] = fmaxf(red[tid], red[tid + s]);
        __syncthreads();
    }
    if (tid == 0) sMax = red[0];
    __syncthreads();
    float off = sMax + 1.0f;
    for (int r = tid; r < NB; r += 1024) {
        float4 ob = make_float4(0.f, 0.f, 0.f, 0.f);
        if (r < NLEV * TOPN) {
            float4 b = sBoxes[r];
            float o = (float)sCls[r] * off;
            ob = make_float4(b.x + o, b.y + o, b.z + o, b.w + o);
        }
        offBoxes[r] = ob;
    }
}

// ------------- IoU 16x16 tiles: async-tensor LDS staging + WMMA area broadcast
__global__ __launch_bounds__(32)
void iou_tiles_k(const float4* __restrict__ offBoxes, unsigned* __restrict__ supMask) {
    int jb = blockIdx.x * 16;            // column (j) base
    int ib = blockIdx.y * 16;            // row (i) base
    if (jb + 15 < ib) return;            // only j > i matters (uniform per block)
    int lane = threadIdx.x;              // one wave32

    __shared__ float4 rowB[16];
    __shared__ float4 colB[16];

    // Stage 16 row boxes + 16 col boxes into LDS with one async-tensor-path op:
    // lanes 0-15 -> rowB, lanes 16-31 -> colB (B128 each, bypasses VGPRs).
    {
        const float4* src = (lane < 16) ? (offBoxes + ib + lane) : (offBoxes + jb + (lane - 16));
        float4* dst = (lane < 16) ? (rowB + lane) : (colB + (lane - 16));
        unsigned ldsOff = (unsigned)(unsigned long long)(void*)dst;  // low 32b of flat addr = LDS offset
        unsigned long long gaddr = (unsigned long long)(const void*)src;
        asm volatile("global_load_async_to_lds_b128 %0, %1, off"
                     :: "v"(ldsOff), "v"(gaddr) : "memory");
        asm volatile("s_wait_asynccnt 0" ::: "memory");
    }
    __syncthreads();

    int l16 = lane & 15;
    float4 rbx = rowB[l16];
    float4 cbx = colB[l16];
    float areaR = fmaxf(rbx.z - rbx.x, 0.0f) * fmaxf(rbx.w - rbx.y, 0.0f);
    float areaC = fmaxf(cbx.z - cbx.x, 0.0f) * fmaxf(cbx.w - cbx.y, 0.0f);

    // sumArea[m][n] = areaR[m] + areaC[n]  as rank-2 outer product on the matrix pipe.
    // A (16x4 f32): lanes 0-15: K0=areaR[M], K1=1 ; lanes 16-31 (K2,K3): 0
    // B (4x16 f32): lanes 0-15: K0=1, K1=areaC[N] ; lanes 16-31 (K2,K3): 0
    bool lo = (lane < 16);
    v2f a, b;
    a.x = lo ? areaR : 0.0f; a.y = lo ? 1.0f : 0.0f;
    b.x = lo ? 1.0f : 0.0f;  b.y = lo ? areaC : 0.0f;
    v8f c = {};
    v8f sumA = __builtin_amdgcn_wmma_f32_16x16x4_f32(false, a, false, b, (short)0, c,
                                                     false, false);

    // C/D layout: VGPR r -> M = r + 8*(lane>=16), N = lane%16. Match it on the VALU.
    float4 myCol = colB[l16];
    int hi8 = (lane >> 4) << 3;
#pragma unroll
    for (int r = 0; r < 8; ++r) {
        float4 rB = rowB[r + hi8];
        float lx = fmaxf(rB.x, myCol.x);
        float ly = fmaxf(rB.y, myCol.y);
        float rx = fminf(rB.z, myCol.z);
        float ry = fminf(rB.w, myCol.w);
        float iw = fmaxf(rx - lx, 0.0f);
        float ih = fmaxf(ry - ly, 0.0f);
        float inter = iw * ih;
        float denom = sumA[r] - inter + 1e-9f;        // > 0 always
        bool sup = inter > 0.6f * denom;              // iou > NMS_TH
        unsigned long long bal = __ballot(sup);       // wave32: bits[15:0]=M=r, bits[31:16]=M=r+8
        if (lane == 0) {
            unsigned lowb  = (unsigned)(bal & 0xFFFFull);
            unsigned highb = (unsigned)((bal >> 16) & 0xFFFFull);
            int word = jb >> 5;
            int sh = jb & 16;
            if (lowb)  atomicOr(&supMask[(ib + r) * MW + word],     lowb  << sh);
            if (highb) atomicOr(&supMask[(ib + r + 8) * MW + word], highb << sh);
        }
    }
}

// -------------------- greedy NMS scan (single wave) + top-100 + clip + output
__global__ __launch_bounds__(32)
void nms_scan_final_k(const unsigned* __restrict__ supMask, const float4* __restrict__ sBoxes,
                      const float* __restrict__ sScore, const int* __restrict__ sCls,
                      const int* __restrict__ sValid, float* __restrict__ out) {
    __shared__ unsigned keep[MW];
    __shared__ int flag;
    __shared__ int list[100];
    int tid = threadIdx.x;

    for (int w = tid; w < MW; w += 32) {
        unsigned bits = 0;
        for (int bix = 0; bix < 32; ++bix) {
            int r = w * 32 + bix;
            if (r < NLEV * TOPN && sValid[r]) bits |= (1u << bix);
        }
        keep[w] = bits;
    }
    __syncthreads();

    for (int i = 0; i < NLEV * TOPN; ++i) {
        if (tid == 0) flag = (keep[i >> 5] >> (i & 31)) & 1;
        __syncthreads();
        if (flag) {
            for (int w = tid; w < MW; w += 32) {
                if (w < (i >> 5)) continue;
                unsigned sup = supMask[i * MW + w];
                if (w == (i >> 5)) {
                    unsigned lowMask = ((i & 31) == 31) ? 0xFFFFFFFFu
                                                        : ((1u << ((i & 31) + 1)) - 1u);
                    sup &= ~lowMask;                  // only suppress j > i
                }
                if (sup) keep[w] &= ~sup;
            }
        }
        __syncthreads();
    }

    if (tid == 0) {
        int cnt = 0;
        for (int r = 0; r < NLEV * TOPN && cnt < 100; ++r)
            if ((keep[r >> 5] >> (r & 31)) & 1) list[cnt++] = r;
        for (; cnt < 100; ++cnt) list[cnt] = -1;
    }
    __syncthreads();

    for (int q = tid; q < 100; q += 32) {
        int t = list[q];
        float bx = 0.f, by = 0.f, bz = 0.f, bw = 0.f, sc = 0.f, cl = 0.f, fv = 0.f;
        if (t >= 0) {
            float4 bb = sBoxes[t];
            float x1 = fminf(fmaxf(bb.x, 0.0f), 2687.0f);
            float y1 = fminf(fmaxf(bb.y, 0.0f), 1599.0f);
            float x2 = fminf(fmaxf(bb.z, 0.0f), 2687.0f);
            float y2 = fminf(fmaxf(bb.w, 0.0f), 1599.0f);
            bool v = (sScore[t] > 0.0f) && (x2 - x1 + 1.0f >= 0.0f) && (y2 - y1 + 1.0f >= 0.0f);
            if (v) { bx = x1; by = y1; bz = x2; bw = y2; sc = sScore[t]; cl = (float)sCls[t]; fv = 1.0f; }
        }
        out[q * 4 + 0] = bx; out[q * 4 + 1] = by; out[q * 4 + 2] = bz; out[q * 4 + 3] = bw;
        out[400 + q] = sc;
        out[500 + q] = cl;
        out[600 + q] = fv;
    }
}

// ================================================================= launch glue
extern "C" void kernel_launch(void* const* d_in, const int* in_sizes, int n_in,
                              void* d_out, int out_size, void* d_ws, size_t ws_size,
                              hipStream_t stream) {
    (void)in_sizes; (void)n_in; (void)out_size; (void)ws_size;
    static const int HW[NLEV] = {67200, 16800, 4200};
    static const long long SOFF[NLEV] = {0LL, 5376000LL, 6720000LL};

    const float* cls[NLEV];  const float* ctr[NLEV];
    const float* scl[NLEV];  const float* sde[NLEV];  const float* box[NLEV];
    for (int l = 0; l < NLEV; ++l) {
        cls[l] = (const float*)d_in[5 * l + 0];
        ctr[l] = (const float*)d_in[5 * l + 1];
        scl[l] = (const float*)d_in[5 * l + 2];
        sde[l] = (const float*)d_in[5 * l + 3];
        box[l] = (const float*)d_in[5 * l + 4];
    }

    // ---- carve workspace (≈31 MB total)
    char* ws = (char*)d_ws;
    size_t off = 0;
    auto carve = [&](size_t bytes) {
        off = (off + 255) & ~(size_t)255;
        void* p = ws + off;
        off += bytes;
        return p;
    };
    float*    scoreBuf  = (float*)   carve(7056000ull * 4);
    unsigned* hist      = (unsigned*)carve((size_t)NLEV * NBINS * 4);
    int*      threshBin = (int*)     carve(NLEV * 4);
    unsigned* candCount = (unsigned*)carve(NLEV * 4);
    float*    candScore = (float*)   carve((size_t)NLEV * CAP * 4);
    int*      candIdx   = (int*)     carve((size_t)NLEV * CAP * 4);
    float*    scandSc   = (float*)   carve((size_t)NLEV * CAP * 4);
    int*      scandIdx  = (int*)     carve((size_t)NLEV * CAP * 4);
    float*    allBoxes  = (float*)   carve(3000ull * 4 * 4);
    float*    allScore  = (float*)   carve(NB * 4);
    int*      allCls    = (int*)     carve(NB * 4);
    int*      allValid  = (int*)     carve(NB * 4);
    float*    gkey      = (float*)   carve(SORT_N * 4);
    int*      gidx      = (int*)     carve(SORT_N * 4);
    float*    sgkey     = (float*)   carve(SORT_N * 4);
    int*      sgidx     = (int*)     carve(SORT_N * 4);
    float4*   sBoxes    = (float4*)  carve((size_t)NB * 16);
    float*    sScore    = (float*)   carve(NB * 4);
    int*      sCls      = (int*)     carve(NB * 4);
    int*      sValid    = (int*)     carve(NB * 4);
    float4*   offBoxes  = (float4*)  carve((size_t)NB * 16);
    unsigned* supMask   = (unsigned*)carve((size_t)NB * MW * 4);

    init_ws_k<<<1024, 256, 0, stream>>>(hist, candCount, candScore, candIdx, supMask);

    for (int l = 0; l < NLEV; ++l) {
        int total = HW[l] * K;
        score_hist_k<<<(total + 255) / 256, 256, 0, stream>>>(
            cls[l], ctr[l], scl[l], scoreBuf + SOFF[l], hist + (size_t)l * NBINS, HW[l]);
    }

    find_thresh_k<<<NLEV, 256, 0, stream>>>(hist, threshBin);

    for (int l = 0; l < NLEV; ++l) {
        int total = HW[l] * K;
        collect_k<<<(total + 255) / 256, 256, 0, stream>>>(
            scoreBuf + SOFF[l], threshBin, candCount, candScore, candIdx, HW[l], l);
    }

    for (int l = 0; l < NLEV; ++l) {
        bitonic4096_k<<<1, 512, 0, stream>>>(candScore + (size_t)l * CAP, candIdx + (size_t)l * CAP,
                                             scandSc + (size_t)l * CAP, scandIdx + (size_t)l * CAP);
    }

    decode_k<<<(NLEV * TOPN + 127) / 128, 128, 0, stream>>>(
        scandSc, scandIdx, sde[0], box[0], sde[1], box[1], sde[2], box[2],
        allBoxes, allScore, allCls, allValid);

    build_gkey_k<<<SORT_N / 256, 256, 0, stream>>>(allScore, allValid, gkey, gidx);
    bitonic4096_k<<<1, 512, 0, stream>>>(gkey, gidx, sgkey, sgidx);

    gather_offset_k<<<1, 1024, 0, stream>>>(sgidx, allBoxes, allScore, allCls, allValid,
                                            sBoxes, sScore, sCls, sValid, offBoxes);

    iou_tiles_k<<<dim3(NB / 16, NB / 16), 32, 0, stream>>>(offBoxes, supMask);

    nms_scan_final_k<<<1, 32, 0, stream>>>(supMask, sBoxes, sScore, sCls, sValid, (float*)d_out);
}